// TransitionProbLayerBeta_21878563406206
// MI455X (gfx1250) — compile-verified
//
#include <hip/hip_runtime.h>

// ---------------------------------------------------------------------------
// TransitionProbLayerBeta for MI455X (gfx1250, wave32).
//
// Shapes: adstock (8192, 2, 366) f32, beta (8,) f32, mu (4,) f32
// Output: (8192, 365, 3, 3) f32  = 107.7 MB  -> pure bandwidth problem.
//
// Per (b,t): O[m] = beta0[m]*a0 + beta1[m]*a1  (m = i*2+j, 4 entries)
//            num[m] = exp(O[m] + mu[m])
//            den_i  = 1 + num[i,0] + num[i,1]
//            Q = [[1-sum(Qd0), Qd00, Qd01],
//                 [Qd10, 1-sum(Qd1), Qd11],
//                 [BASIS, BASIS, 1.0]]          (all max'd with BASIS)
//
// The K=2 einsum is embedded in one V_WMMA_F32_16X16X4_F32 per wave:
//   A (16x4): rows 0..3 hold beta over K=0,1 ; rows 8..11 hold beta over K=2,3
//             -> in lane terms: lanes 0..3 and 24..27 carry beta, rest zero.
//   B (4x16): every lane writes its own (a0,a1) into the two B VGPRs.
//             Lanes 0..15 thereby populate K=0,1 of columns 0..15 (their own
//             element) and lanes 16..31 populate K=2,3 of the same columns
//             (their own element).
//   D       : lane L receives its own element's O[m] in D VGPRs 0..3
//             (rows 0..3 for lanes 0..15, rows 8..11 for lanes 16..31).
// Because A and B share the same (lane,VGPR)->K mapping and cross terms are
// zero, the result is independent of the exact K ordering in the layout.
// Zero cross-lane traffic for operands or results.
// ---------------------------------------------------------------------------

typedef __attribute__((ext_vector_type(2))) float v2f;
typedef __attribute__((ext_vector_type(8))) float v8f;

#define T_DIM   365
#define B_DIM   8192
#define ROW_LEN 366               // adstock innermost length (T+1)
#define TOTAL   (B_DIM * T_DIM)   // 2,990,080 = 11680 * 256 exactly
#define BASIS_F 1e-10f

__global__ __launch_bounds__(256) void tpl_beta_kernel(
    const float* __restrict__ adstock,
    const float* __restrict__ beta,
    const float* __restrict__ mu,
    float* __restrict__ out)
{
    __shared__ float smem[8 * 288];   // 8 waves * 288 dwords = 9216 B

    const int tid  = threadIdx.x;
    const int lane = tid & 31;
    const int wave = tid >> 5;
    const int g    = blockIdx.x * 256 + tid;   // site index, always < TOTAL
    const int b    = g / T_DIM;
    const int t    = g - b * T_DIM;

    // ---- B matrix: this lane's own two adstock channels (coalesced loads) --
    const float* p = adstock + b * (2 * ROW_LEN) + (t + 1);
    v2f Bm;
    Bm[0] = p[0];        // c = 0
    Bm[1] = p[ROW_LEN];  // c = 1

    // ---- A matrix: beta on rows 0..3 (K=0,1) and rows 8..11 (K=2,3) -------
    // Lane view: lanes 0..3 -> rows 0..3, lanes 24..27 -> rows 8..11.
    const int  i3  = lane & 3;
    const bool sel = (lane < 4) || ((unsigned)(lane - 24) < 4u);
    const float be0 = beta[i3];       // beta_r[c=0, m]  (broadcast load)
    const float be1 = beta[4 + i3];   // beta_r[c=1, m]
    v2f Am;
    Am[0] = sel ? be0 : 0.0f;
    Am[1] = sel ? be1 : 0.0f;

    // ---- WMMA: D[m, n] = sum_K A[m,K] * B[K,n] ----------------------------
    // EXEC is all-ones here: grid is an exact multiple of the block size and
    // there is no divergent control flow before this point.
    v8f C = {0.f, 0.f, 0.f, 0.f, 0.f, 0.f, 0.f, 0.f};
    v8f D = __builtin_amdgcn_wmma_f32_16x16x4_f32(
        /*neg_a=*/false, Am, /*neg_b=*/false, Bm,
        /*c_mod=*/(short)0, C, /*reuse_a=*/false, /*reuse_b=*/false);

    // Every lane now holds its own O_beta[m], m=0..3, in D[0..3].
    const float n0 = expf(D[0] + mu[0]);   // num[i=0,j=0]
    const float n1 = expf(D[1] + mu[1]);   // num[i=0,j=1]
    const float n2 = expf(D[2] + mu[2]);   // num[i=1,j=0]
    const float n3 = expf(D[3] + mu[3]);   // num[i=1,j=1]

    const float inv0 = 1.0f / (1.0f + n0 + n1);
    const float inv1 = 1.0f / (1.0f + n2 + n3);
    const float qd00 = n0 * inv0, qd01 = n1 * inv0;
    const float qd10 = n2 * inv1, qd11 = n3 * inv1;
    const float qs0  = 1.0f - (qd00 + qd01);   // match reference's formulation
    const float qs1  = 1.0f - (qd10 + qd11);

    float q[9];
    q[0] = fmaxf(qs0,  BASIS_F);
    q[1] = fmaxf(qd00, BASIS_F);
    q[2] = fmaxf(qd01, BASIS_F);
    q[3] = fmaxf(qd10, BASIS_F);
    q[4] = fmaxf(qs1,  BASIS_F);
    q[5] = fmaxf(qd11, BASIS_F);
    q[6] = BASIS_F;
    q[7] = BASIS_F;
    q[8] = 1.0f;   // 1 - 2e-10 rounds to 1.0f in fp32 (matches reference)

    // ---- LDS re-tile: 36B/lane strided -> fully coalesced wave stores -----
    // Write stride = 9 dwords; gcd(9,64)=1 -> bank-conflict-free.
    float* slab = smem + wave * 288;
#pragma unroll
    for (int m = 0; m < 9; ++m)
        slab[lane * 9 + m] = q[m];

    __syncthreads();

    // Wave covers out dwords [wave_g0*9, wave_g0*9 + 288): contiguous.
    const int wave_g0 = blockIdx.x * 256 + wave * 32;
    float* dst = out + (size_t)wave_g0 * 9;
#pragma unroll
    for (int k = 0; k < 9; ++k)
        dst[k * 32 + lane] = slab[k * 32 + lane];
}

extern "C" void kernel_launch(void* const* d_in, const int* in_sizes, int n_in,
                              void* d_out, int out_size, void* d_ws, size_t ws_size,
                              hipStream_t stream) {
    const float* adstock = (const float*)d_in[0];  // (8192, 2, 366)
    const float* beta    = (const float*)d_in[1];  // (8,)
    const float* mu      = (const float*)d_in[2];  // (4,)
    float* out           = (float*)d_out;          // (8192, 365, 3, 3)

    (void)in_sizes; (void)n_in; (void)out_size; (void)d_ws; (void)ws_size;

    const int block = 256;
    const int grid  = TOTAL / block;  // 11680, exact
    tpl_beta_kernel<<<grid, block, 0, stream>>>(adstock, beta, mu, out);
}